// GCNBlock_11914239279898
// MI455X (gfx1250) — compile-verified
//
#include <hip/hip_runtime.h>
#include <hip/hip_bf16.h>

typedef __attribute__((ext_vector_type(2))) float v2f;
typedef __attribute__((ext_vector_type(8))) float v8f;

#define IN_DIM 128
#define OUT_DIM 256
#define LN_EPS 1e-5f

// ---------------- zero workspace ----------------
__global__ void zero_u32(unsigned* __restrict__ p, unsigned n) {
    unsigned i = blockIdx.x * blockDim.x + threadIdx.x;
    if (i < n) p[i] = 0u;
}

// ---------------- degree counting ----------------
__global__ void degree_kernel(const int* __restrict__ src, const int* __restrict__ dst,
                              int* __restrict__ deg_out, int* __restrict__ deg_in, int n_edges) {
    int e = blockIdx.x * blockDim.x + threadIdx.x;
    if (e >= n_edges) return;
    atomicAdd(&deg_out[src[e]], 1);
    atomicAdd(&deg_in[dst[e]], 1);
}

// ---------------- h = features * rsqrt(max(deg_out,1)) ----------------
__global__ void compute_h(const float* __restrict__ feat, const int* __restrict__ deg_out,
                          float* __restrict__ h, int n_nodes) {
    int idx = blockIdx.x * blockDim.x + threadIdx.x;    // one float4 per thread
    int total = n_nodes * (IN_DIM / 4);
    if (idx >= total) return;
    int node = idx / (IN_DIM / 4);
    float nrm = rsqrtf(fmaxf((float)deg_out[node], 1.0f));
    float4 v = ((const float4*)feat)[idx];
    v.x *= nrm; v.y *= nrm; v.z *= nrm; v.w *= nrm;
    ((float4*)h)[idx] = v;
}

// ---------------- SpMM scatter: agg[dst] += h[src] ----------------
__global__ __launch_bounds__(256) void spmm_scatter(const int* __restrict__ src,
                                                    const int* __restrict__ dst,
                                                    const float* __restrict__ h,
                                                    float* __restrict__ agg, int n_edges) {
    int edge = blockIdx.x * (blockDim.x >> 5) + (threadIdx.x >> 5);
    if (edge >= n_edges) return;
    int lane = threadIdx.x & 31;
    int s = src[edge];
    int d = dst[edge];
    const float4* hp = (const float4*)(h + (size_t)s * IN_DIM);
    float4 v = hp[lane];                        // 32 lanes * 4 = 128 floats
    float* ap = agg + (size_t)d * IN_DIM + lane * 4;
    __hip_atomic_fetch_add(ap + 0, v.x, __ATOMIC_RELAXED, __HIP_MEMORY_SCOPE_AGENT);
    __hip_atomic_fetch_add(ap + 1, v.y, __ATOMIC_RELAXED, __HIP_MEMORY_SCOPE_AGENT);
    __hip_atomic_fetch_add(ap + 2, v.z, __ATOMIC_RELAXED, __HIP_MEMORY_SCOPE_AGENT);
    __hip_atomic_fetch_add(ap + 3, v.w, __ATOMIC_RELAXED, __HIP_MEMORY_SCOPE_AGENT);
}

// ---------------- fused: (agg@W)*norm_dst + b -> LN -> ReLU, + feat@skip_W + skip_b ----
// One block = 16 rows x 256 cols; 8 waves, each owns two 16-wide N-tiles for BOTH
// GEMMs. K outermost so A fragments (k-only dependent) are loaded once per step
// and feed 4 v_wmma_f32_16x16x4_f32. norm_dst is applied to the accumulators in
// the epilogue (diag-scale commutes with the GEMM), keeping the K-loop pure
// load+WMMA.
__global__ __launch_bounds__(256) void gcn_fused(const float* __restrict__ agg,
                                                 const int* __restrict__ deg_in,
                                                 const float* __restrict__ W,
                                                 const float* __restrict__ bias,
                                                 const float* __restrict__ gamma,
                                                 const float* __restrict__ beta,
                                                 const float* __restrict__ feat,
                                                 const float* __restrict__ skip_W,
                                                 const float* __restrict__ skip_b,
                                                 float* __restrict__ out, int n_nodes) {
    __shared__ float lds_gcn[16][OUT_DIM + 4];
    __shared__ float lds_skip[16][OUT_DIM + 4];

    const int m0   = blockIdx.x * 16;
    const int wave = threadIdx.x >> 5;      // 0..7
    const int lane = threadIdx.x & 31;
    const int half = lane >> 4;             // 0 or 1 (K-pair select)
    const int l16  = lane & 15;

    // A-matrix row for this lane (ISA 16x4 f32 A layout)
    const int arow = m0 + l16;
    const float* agg_row  = agg  + (size_t)arow * IN_DIM;
    const float* feat_row = feat + (size_t)arow * IN_DIM;
    const int bcol = wave * 32 + l16;       // first N-tile column for this lane

    v8f cg0 = {}, cg1 = {};                 // agg@W accumulators (tiles t=0,1)
    v8f cs0 = {}, cs1 = {};                 // feat@skip_W accumulators

    for (int k0 = 0; k0 < IN_DIM; k0 += 4) {
        const int ka = k0 + 2 * half;
        // A fragments: shared by both N-tiles
        v2f aA = *(const v2f*)(agg_row + ka);
        v2f aF = *(const v2f*)(feat_row + ka);
        // B fragments: two K rows x two N-tiles, both weight matrices
        const float* Wk0 = W      + (size_t)ka * OUT_DIM;
        const float* Wk1 = Wk0 + OUT_DIM;
        const float* Sk0 = skip_W + (size_t)ka * OUT_DIM;
        const float* Sk1 = Sk0 + OUT_DIM;
        v2f bW0, bW1, bS0, bS1;
        bW0[0] = Wk0[bcol];      bW0[1] = Wk1[bcol];
        bW1[0] = Wk0[bcol + 16]; bW1[1] = Wk1[bcol + 16];
        bS0[0] = Sk0[bcol];      bS0[1] = Sk1[bcol];
        bS1[0] = Sk0[bcol + 16]; bS1[1] = Sk1[bcol + 16];
        cg0 = __builtin_amdgcn_wmma_f32_16x16x4_f32(false, aA, false, bW0, (short)0, cg0, false, false);
        cg1 = __builtin_amdgcn_wmma_f32_16x16x4_f32(false, aA, false, bW1, (short)0, cg1, false, false);
        cs0 = __builtin_amdgcn_wmma_f32_16x16x4_f32(false, aF, false, bS0, (short)0, cs0, false, false);
        cs1 = __builtin_amdgcn_wmma_f32_16x16x4_f32(false, aF, false, bS1, (short)0, cs1, false, false);
    }

    // Per-row norm_dst for the 8 accumulator rows this lane holds
    float nrmv[8];
#pragma unroll
    for (int i = 0; i < 8; ++i) {
        const int row = i + 8 * half;
        nrmv[i] = rsqrtf(fmaxf((float)deg_in[m0 + row], 1.0f));
    }

    // Spill C/D tiles to LDS: gcn scaled by norm_dst + bias, skip + skip_b
    {
        const float bb0 = bias[bcol],        sb0 = skip_b[bcol];
        const float bb1 = bias[bcol + 16],   sb1 = skip_b[bcol + 16];
#pragma unroll
        for (int i = 0; i < 8; ++i) {
            const int row = i + 8 * half;
            lds_gcn[row][bcol]       = cg0[i] * nrmv[i] + bb0;
            lds_gcn[row][bcol + 16]  = cg1[i] * nrmv[i] + bb1;
            lds_skip[row][bcol]      = cs0[i] + sb0;
            lds_skip[row][bcol + 16] = cs1[i] + sb1;
        }
    }
    __syncthreads();

    // LayerNorm over the 256-wide rows; each wave handles 2 rows.
#pragma unroll
    for (int rr = 0; rr < 2; ++rr) {
        const int row = wave * 2 + rr;
        float vals[8];
        float s = 0.f, s2 = 0.f;
#pragma unroll
        for (int j = 0; j < 8; ++j) {
            float x = lds_gcn[row][lane * 8 + j];
            vals[j] = x;
            s += x;
            s2 += x * x;
        }
        for (int off = 16; off > 0; off >>= 1) {
            s  += __shfl_xor(s,  off, 32);
            s2 += __shfl_xor(s2, off, 32);
        }
        const float mu  = s * (1.0f / OUT_DIM);
        const float var = s2 * (1.0f / OUT_DIM) - mu * mu;
        const float rs  = rsqrtf(var + LN_EPS);
        float* orow = out + (size_t)(m0 + row) * OUT_DIM;
#pragma unroll
        for (int j = 0; j < 8; ++j) {
            const int c = lane * 8 + j;
            float ln = (vals[j] - mu) * rs * gamma[c] + beta[c];
            orow[c] = fmaxf(ln, 0.f) + lds_skip[row][c];
        }
    }
}

extern "C" void kernel_launch(void* const* d_in, const int* in_sizes, int n_in,
                              void* d_out, int out_size, void* d_ws, size_t ws_size,
                              hipStream_t stream) {
    const float* feat   = (const float*)d_in[0];
    const int*   src    = (const int*)d_in[1];
    const int*   dst    = (const int*)d_in[2];
    const float* W      = (const float*)d_in[3];
    const float* bias   = (const float*)d_in[4];
    const float* gamma  = (const float*)d_in[5];
    const float* beta   = (const float*)d_in[6];
    const float* skip_W = (const float*)d_in[7];
    const float* skip_b = (const float*)d_in[8];
    float* out = (float*)d_out;

    const int n_nodes = in_sizes[0] / IN_DIM;
    const int n_edges = in_sizes[1];

    // Workspace layout: deg_out[N] (int) | deg_in[N] (int) | h[N*128] | agg[N*128]
    int*   deg_out = (int*)d_ws;
    int*   deg_in  = deg_out + n_nodes;
    float* h       = (float*)(deg_in + n_nodes);
    float* agg     = h + (size_t)n_nodes * IN_DIM;

    const int T = 256;

    // Zero degree counters and agg accumulator (every call: graph replay safe)
    {
        unsigned n = 2u * (unsigned)n_nodes;
        zero_u32<<<(n + T - 1) / T, T, 0, stream>>>((unsigned*)deg_out, n);
        unsigned na = (unsigned)n_nodes * IN_DIM;
        zero_u32<<<(na + T - 1) / T, T, 0, stream>>>((unsigned*)agg, na);
    }

    degree_kernel<<<(n_edges + T - 1) / T, T, 0, stream>>>(src, dst, deg_out, deg_in, n_edges);

    {
        int total = n_nodes * (IN_DIM / 4);
        compute_h<<<(total + T - 1) / T, T, 0, stream>>>(feat, deg_out, h, n_nodes);
    }

    {
        int waves_per_block = T / 32;
        int blocks = (n_edges + waves_per_block - 1) / waves_per_block;
        spmm_scatter<<<blocks, T, 0, stream>>>(src, dst, h, agg, n_edges);
    }

    gcn_fused<<<n_nodes / 16, T, 0, stream>>>(agg, deg_in, W, bias, gamma, beta,
                                              feat, skip_W, skip_b, out, n_nodes);
}